// Viterbi_24361054503465
// MI455X (gfx1250) — compile-verified
//
#include <hip/hip_runtime.h>
#include <stdint.h>

#define T_LEN  1024
#define BATCH  256
#define NST    128
#define TRS    132          // padded row stride (dwords): 4*jj mod 64 tiles banks optimally
#define START_I 126
#define END_I   127
#define NEGV   (-10000.0f)

// Low 32 bits of a generic pointer into LDS == wave-relative LDS byte offset
// (ISA 10.2: LDS aperture maps flat addr[31:0] -> LDS address).
__device__ __forceinline__ unsigned lds_off32(const void* p) {
  return (unsigned)(uintptr_t)p;
}

// CDNA5 async data mover: copy 16B/lane from global to LDS, tracked by ASYNCcnt.
__device__ __forceinline__ void async_g2l_b128(unsigned lds_dst, const void* gsrc) {
  asm volatile("global_load_async_to_lds_b128 %0, %1, off"
               :: "v"(lds_dst), "v"((unsigned long long)(uintptr_t)gsrc)
               : "memory");
}
__device__ __forceinline__ void wait_async0() {
  asm volatile("s_wait_asynccnt 0" ::: "memory");
}

__global__ __launch_bounds__(128)
void viterbi_fwd_bwd(const float* __restrict__ unary,
                     const float* __restrict__ trans,
                     const int*   __restrict__ lengths,
                     int*   __restrict__ path_out,
                     float* __restrict__ score_out,
                     unsigned char* __restrict__ bp_ws)
{
  __shared__ __align__(16) float trP[NST * TRS];   // trP[j*132 + k] = trans[j*128 + k]
  __shared__ __align__(16) float alpha[2][NST];    // double buffer -> 1 barrier per step
  __shared__ __align__(16) float ubuf[2][NST];     // async double buffer for unary rows
  __shared__ __align__(16) float pm[NST];
  __shared__ __align__(16) int   pi[NST];
  __shared__ int path_lds[T_LEN];

  const int tid = threadIdx.x;          // == state index j, 0..127
  const int b   = blockIdx.x;
  const int len = lengths[b];

  // Stage transition matrix rows into padded LDS layout (float4 both sides).
  for (int idx = tid * 4; idx < NST * NST; idx += 128 * 4) {
    const float4 v = *(const float4*)&trans[idx];
    const int j = idx >> 7, k = idx & (NST - 1);
    *(float4*)&trP[j * TRS + k] = v;               // 132*4=528 B rows: 16B-aligned
  }
  alpha[0][tid] = (tid == START_I) ? 0.0f : NEGV;

  // Prefetch unary row t=0 via async data mover (wave 0 only, 32 lanes x 16B = 512B).
  if (tid < 32) {
    async_g2l_b128(lds_off32(&ubuf[0][0]) + (unsigned)tid * 16u,
                   (const char*)(unary + (size_t)b * NST) + tid * 16);
    wait_async0();
  }
  __syncthreads();

  const float* tp = &trP[tid * TRS];               // this thread's full row (128 k)

  for (int t = 0; t < T_LEN; ++t) {
    const int cur = t & 1, nxt = cur ^ 1;
    // Kick off async copy of next unary row behind this step's compute.
    if (tid < 32 && (t + 1) < T_LEN) {
      async_g2l_b128(lds_off32(&ubuf[nxt][0]) + (unsigned)tid * 16u,
                     (const char*)(unary + ((size_t)(t + 1) * BATCH + b) * NST) + tid * 16);
    }

    // 4 independent (max, argmax) chains over residues k mod 4 -> 4-way ILP.
    // Strict '>' inside a chain keeps first-occurrence within the residue class.
    const float* ac = &alpha[cur][0];
    float m0 = -3.4e38f, m1 = -3.4e38f, m2 = -3.4e38f, m3 = -3.4e38f;
    int   i0 = 0, i1 = 0, i2 = 0, i3 = 0;
#pragma unroll 8
    for (int k = 0; k < NST; k += 4) {
      const float4 a4 = *(const float4*)&ac[k];   // uniform ds_load_b128 (broadcast)
      const float4 t4 = *(const float4*)&tp[k];   // per-lane ds_load_b128, optimal banks
      const float s0 = a4.x + t4.x;
      const float s1 = a4.y + t4.y;
      const float s2 = a4.z + t4.z;
      const float s3 = a4.w + t4.w;
      if (s0 > m0) { m0 = s0; i0 = k + 0; }
      if (s1 > m1) { m1 = s1; i1 = k + 1; }
      if (s2 > m2) { m2 = s2; i2 = k + 2; }
      if (s3 > m3) { m3 = s3; i3 = k + 3; }
    }
    // Merge chains; on equal value prefer smaller k (jnp.argmax first-occurrence).
    if (m1 > m0 || (m1 == m0 && i1 < i0)) { m0 = m1; i0 = i1; }
    if (m3 > m2 || (m3 == m2 && i3 < i2)) { m2 = m3; i2 = i3; }
    if (m2 > m0 || (m2 == m0 && i2 < i0)) { m0 = m2; i0 = i2; }

    // Sole writer of alpha[nxt][tid]; nobody reads alpha[nxt] this step.
    const float old = alpha[cur][tid];
    const volatile float* uv = &ubuf[cur][0];     // written only by async engine
    const float na = m0 + uv[tid];
    alpha[nxt][tid] = (t < len) ? na : old;
    bp_ws[((size_t)t * BATCH + b) * NST + tid] = (unsigned char)i0;

    if (tid < 32) wait_async0();                  // next-row LDS writes landed
    __syncthreads();                              // publish alpha[nxt] + ubuf[nxt]
  }

  // Terminal: score and argmax of alpha_final + tr[END, :]  (T_LEN even -> buffer 0)
  pm[tid] = alpha[T_LEN & 1][tid] + trans[END_I * NST + tid];
  pi[tid] = tid;
  __threadfence();                                // bp stores visible to this block's loads
  __syncthreads();
  for (int off = 64; off >= 1; off >>= 1) {
    if (tid < off) {
      const float v1 = pm[tid], v2 = pm[tid + off];
      const int   j1 = pi[tid], j2 = pi[tid + off];
      if (v2 > v1 || (v2 == v1 && j2 < j1)) { pm[tid] = v2; pi[tid] = j2; }
    }
    __syncthreads();
  }

  // Backtrace (serial chain per batch; bp is L2-resident u8).
  if (tid == 0) {
    score_out[b] = pm[0];
    int best = pi[0];
    for (int t = T_LEN - 1; t >= 0; --t) {
      if (t < len) {
        path_lds[t] = best;
        best = (int)bp_ws[((size_t)t * BATCH + b) * NST + best];
      } else {
        path_lds[t] = 0;
      }
    }
  }
  __syncthreads();
  for (int t = tid; t < T_LEN; t += 128) {
    path_out[(size_t)t * BATCH + b] = path_lds[t];
  }
}

extern "C" void kernel_launch(void* const* d_in, const int* in_sizes, int n_in,
                              void* d_out, int out_size, void* d_ws, size_t ws_size,
                              hipStream_t stream) {
  const float* unary   = (const float*)d_in[0];   // (T,B,N) f32
  const float* trans   = (const float*)d_in[1];   // (1,N,N) f32
  const int*   lengths = (const int*)d_in[2];     // (B,) i32

  int*   path_out  = (int*)d_out;                                   // (T,B) i32
  float* score_out = (float*)((int*)d_out + (size_t)T_LEN * BATCH); // (B,) f32
  unsigned char* bp_ws = (unsigned char*)d_ws;                      // T*B*N u8 = 32 MiB

  viterbi_fwd_bwd<<<BATCH, 128, 0, stream>>>(unary, trans, lengths,
                                             path_out, score_out, bp_ws);
}